// MomentHead_26594437497382
// MI455X (gfx1250) — compile-verified
//
#include <hip/hip_runtime.h>
#include <hip/hip_bf16.h>
#include <math.h>

#define EPSF 1e-05f
#define BN_EPSF 1e-05f

typedef __attribute__((ext_vector_type(2))) float v2f;
typedef __attribute__((ext_vector_type(8))) float v8f;

__device__ __forceinline__ v8f wmma4(v2f a, v2f b, v8f c)
{
    return __builtin_amdgcn_wmma_f32_16x16x4_f32(false, a, false, b,
                                                 (short)0, c, false, false);
}

// ---------------------------------------------------------------------------
// Batched GEMM: C[b] = alpha * op(A[b]) @ B[b] (+ bias[col])
// op(A)=A (MxK row-major, lda) or A^T when TRANSA=1 ((m,k) at A[k*lda+m]).
// One wave computes a 32x64 tile (2 row-halves x 4 N-tiles, 8 WMMAs / K-step)
// with V_WMMA_F32_16X16X4_F32. Block = 8 waves = 128x128 tile.
// REQUIREMENT (holds for all launches here): N % 128 == 0, K % 4 == 0.
// Row edges handled branch-free via clamped addresses + 0/1 mask multiply.
// ---------------------------------------------------------------------------
template <int TRANSA>
__global__ void __launch_bounds__(256)
gemm_f32_wmma(const float* __restrict__ A, const float* __restrict__ B,
              float* __restrict__ C, const float* __restrict__ bias,
              int M, int N, int K,
              int lda, int ldb, int ldc,
              long long sA, long long sB, long long sC,
              float alpha)
{
    const int bz = blockIdx.z;
    const float* Ab = A + (long long)bz * sA;
    const float* Bb = B + (long long)bz * sB;
    float*       Cb = C + (long long)bz * sC;

    const int lane = threadIdx.x & 31;
    const int wave = threadIdx.x >> 5;
    const int m0 = blockIdx.y * 128 + (wave & 3) * 32;
    if (m0 >= M) return;                       // wave-uniform exit
    const int n0 = blockIdx.x * 128 + (wave >> 2) * 64;

    // A fragment (ISA 7.12.2, 32-bit A 16x4): lane L -> row m0+L%16,
    // K pair = 2*(L/16) + {0,1}
    const int lm   = lane & 15;
    const int koff = (lane >> 4) << 1;         // 0 or 2

    const int mr0 = m0 + lm;
    const int mr1 = m0 + 16 + lm;
    const float msk0 = (mr0 < M) ? 1.0f : 0.0f;
    const float msk1 = (mr1 < M) ? 1.0f : 0.0f;
    const int mc0 = (mr0 < M) ? mr0 : 0;       // clamped safe row addresses
    const int mc1 = (mr1 < M) ? mr1 : 0;

    // B column pointers (K-invariant); lanes 0..15 cover 64B contiguous.
    const float* bp0 = Bb + (n0 + 0 * 16 + lm);
    const float* bp1 = Bb + (n0 + 1 * 16 + lm);
    const float* bp2 = Bb + (n0 + 2 * 16 + lm);
    const float* bp3 = Bb + (n0 + 3 * 16 + lm);

    v8f acc0[4] = {};
    v8f acc1[4] = {};

#pragma unroll 2
    for (int k = 0; k < K; k += 4) {
        const int ka = k + koff;
        v2f a0, a1;
        if (TRANSA == 0) {
            const float* p0 = Ab + (long long)mc0 * lda + ka;
            const float* p1 = Ab + (long long)mc1 * lda + ka;
            a0.x = p0[0]; a0.y = p0[1];
            a1.x = p1[0]; a1.y = p1[1];
        } else {
            const float* pk  = Ab + (long long)ka * lda;
            const float* pk1 = Ab + (long long)(ka + 1) * lda;
            a0.x = pk[mc0]; a0.y = pk1[mc0];
            a1.x = pk[mc1]; a1.y = pk1[mc1];
        }
        a0.x *= msk0; a0.y *= msk0;
        a1.x *= msk1; a1.y *= msk1;

        const long long r0 = (long long)ka * ldb;
        const long long r1 = (long long)(ka + 1) * ldb;
        v2f b0, b1, b2, b3;
        b0.x = bp0[r0]; b0.y = bp0[r1];
        b1.x = bp1[r0]; b1.y = bp1[r1];
        b2.x = bp2[r0]; b2.y = bp2[r1];
        b3.x = bp3[r0]; b3.y = bp3[r1];

        acc0[0] = wmma4(a0, b0, acc0[0]);
        acc0[1] = wmma4(a0, b1, acc0[1]);
        acc0[2] = wmma4(a0, b2, acc0[2]);
        acc0[3] = wmma4(a0, b3, acc0[3]);
        acc1[0] = wmma4(a1, b0, acc1[0]);
        acc1[1] = wmma4(a1, b1, acc1[1]);
        acc1[2] = wmma4(a1, b2, acc1[2]);
        acc1[3] = wmma4(a1, b3, acc1[3]);
    }

    // C/D layout: VGPR r -> row base+r (lanes 0-15), base+r+8 (lanes 16-31)
    const int rbase = (lane >> 4) * 8;
#pragma unroll
    for (int t = 0; t < 4; ++t) {
        const int col = n0 + t * 16 + lm;      // always < N here
        const float badd = bias ? bias[col] : 0.0f;
#pragma unroll
        for (int r = 0; r < 8; ++r) {
            const int row0 = m0 + rbase + r;
            if (row0 < M)
                Cb[(long long)row0 * ldc + col] = alpha * acc0[t][r] + badd;
        }
#pragma unroll
        for (int r = 0; r < 8; ++r) {
            const int row1 = m0 + 16 + rbase + r;
            if (row1 < M)
                Cb[(long long)row1 * ldc + col] = alpha * acc1[t][r] + badd;
        }
    }
}

// ---------------------------------------------------------------------------
// Small support kernels
// ---------------------------------------------------------------------------
__device__ __forceinline__ float block_reduce_256(float v, float* red)
{
    red[threadIdx.x] = v;
    __syncthreads();
    for (int off = 128; off > 0; off >>= 1) {
        if ((int)threadIdx.x < off) red[threadIdx.x] += red[threadIdx.x + off];
        __syncthreads();
    }
    return red[0];
}

__global__ void deg_kernel(const float* __restrict__ graph,
                           float* __restrict__ inv_sqrt, int N)
{
    __shared__ float red[256];
    const int bn = blockIdx.x;
    const float* row = graph + (long long)bn * N;
    float s = 0.0f;
    for (int m = threadIdx.x; m < N; m += 256) s += row[m];
    float tot = block_reduce_256(s, red);
    if (threadIdx.x == 0) {
        float deg = tot < EPSF ? EPSF : tot;
        inv_sqrt[bn] = 1.0f / sqrtf(deg);
    }
}

__global__ void w_rowsum_kernel(const float* __restrict__ graph,
                                const float* __restrict__ invs,
                                float* __restrict__ W,
                                float* __restrict__ row_sum, int N)
{
    __shared__ float red[256];
    const int bn = blockIdx.x;
    const int b = bn / N;
    const float* g   = graph + (long long)bn * N;
    const float* isb = invs + (long long)b * N;
    float* Wrow = W + (long long)bn * N;
    const float isn = invs[bn];
    float s = 0.0f;
    for (int m = threadIdx.x; m < N; m += 256) {
        float w = g[m] * isn * isb[m];
        Wrow[m] = w;
        s += w;
    }
    float tot = block_reduce_256(s, red);
    if (threadIdx.x == 0) row_sum[bn] = tot;
}

__global__ void trace_kernel(const float* __restrict__ Mats,
                             float* __restrict__ tr, int D)
{
    __shared__ float red[256];
    const int b = blockIdx.x;
    const float* Mb = Mats + (long long)b * D * D;
    float s = 0.0f;
    for (int i = threadIdx.x; i < D; i += 256) s += Mb[(long long)i * D + i];
    float tot = block_reduce_256(s, red);
    if (threadIdx.x == 0) tr[b] = tot;
}

__global__ void mu_wc_kernel(const float* __restrict__ tokens,
                             const float* __restrict__ row_sum,
                             const float* __restrict__ tr_w,
                             float* __restrict__ mu, float* __restrict__ wc,
                             int N, int D)
{
    const int b = blockIdx.x;
    const int d = threadIdx.x;
    if (d >= D) return;
    const float* tb = tokens + (long long)b * N * D;
    const float* rs = row_sum + (long long)b * N;
    float s = 0.0f, srs = 0.0f;
    for (int n = 0; n < N; ++n) {
        float r = rs[n];
        s += tb[(long long)n * D + d] * r;
        srs += r;
    }
    const float itr = 1.0f / (tr_w[b] + EPSF);
    const float m = s * itr;
    mu[(long long)b * D + d] = m;
    wc[(long long)b * D + d] = m * (1.0f - srs * itr);
}

__global__ void xc_kernel(const float* __restrict__ tokens,
                          const float* __restrict__ mu,
                          float* __restrict__ Xc, int N, int D)
{
    const int bn = blockIdx.x;
    const int b = bn / N;
    const int d = threadIdx.x;
    if (d >= D) return;
    const long long o = (long long)bn * D + d;
    Xc[o] = tokens[o] - mu[(long long)b * D + d];
}

__global__ void ns_init_kernel(float* __restrict__ Z, float* __restrict__ Y,
                               const float* __restrict__ tr, int D)
{
    const int b = blockIdx.y;
    const int idx = blockIdx.x * 256 + threadIdx.x;
    if (idx >= D * D) return;
    const long long o = (long long)b * D * D + idx;
    const float itr = 1.0f / (tr[b] + EPSF);
    Z[o] *= itr;
    const int i = idx / D, j = idx % D;
    Y[o] = (i == j) ? 1.0f : 0.0f;
}

__global__ void neg3i_kernel(float* __restrict__ P, int D)
{
    const int b = blockIdx.y;
    const int idx = blockIdx.x * 256 + threadIdx.x;
    if (idx >= D * D) return;
    const long long o = (long long)b * D * D + idx;
    const int i = idx / D, j = idx % D;
    float v = -P[o];
    if (i == j) v += 3.0f;
    P[o] = v;
}

__global__ void triu_kernel(const float* __restrict__ Y,
                            const float* __restrict__ trM2,
                            float* __restrict__ vec, int D, int TRI)
{
    const int b = blockIdx.y;
    const int t = blockIdx.x * 256 + threadIdx.x;
    if (t >= TRI) return;
    double disc = (double)(2 * D + 1) * (2 * D + 1) - 8.0 * (double)t;
    int i = (int)(((double)(2 * D + 1) - sqrt(disc)) * 0.5);
    if (i > D - 1) i = D - 1;
    if (i < 0) i = 0;
    while (i > 0 && (long long)i * D - (long long)i * (i - 1) / 2 > t) --i;
    while ((long long)(i + 1) * D - (long long)(i + 1) * i / 2 <= t) ++i;
    const long long base = (long long)i * D - (long long)i * (i - 1) / 2;
    const int j = i + (int)(t - base);
    const float scale = 1.0f / sqrtf(trM2[b] + EPSF);
    vec[(long long)b * TRI + t] =
        Y[(long long)b * D * D + (long long)i * D + j] * scale;
}

__global__ void bn_gelu_kernel(float* __restrict__ h,
                               const float* __restrict__ gamma,
                               const float* __restrict__ beta,
                               const float* __restrict__ rm,
                               const float* __restrict__ rv,
                               int H, long long total)
{
    const long long idx = (long long)blockIdx.x * 256 + threadIdx.x;
    if (idx >= total) return;
    const int j = (int)(idx % H);
    float x = h[idx];
    x = (x - rm[j]) * (1.0f / sqrtf(rv[j] + BN_EPSF)) * gamma[j] + beta[j];
    h[idx] = 0.5f * x * (1.0f + erff(x * 0.70710678118654752440f));
}

__global__ void zero_kernel(float* __restrict__ p, long long n)
{
    const long long idx = (long long)blockIdx.x * 256 + threadIdx.x;
    if (idx < n) p[idx] = 0.0f;
}

__global__ void sketch_scatter_kernel(const float* __restrict__ wc,
                                      const int* __restrict__ h1,
                                      const int* __restrict__ h2,
                                      const int* __restrict__ h3,
                                      const float* __restrict__ s1,
                                      const float* __restrict__ s2,
                                      const float* __restrict__ s3,
                                      float* __restrict__ o1,
                                      float* __restrict__ o2,
                                      float* __restrict__ o3,
                                      int D, int S)
{
    const int b = blockIdx.x;
    const int d = threadIdx.x;
    if (d >= D) return;
    const float v = wc[(long long)b * D + d];
    atomicAdd(&o1[(long long)b * S + h1[d]], v * s1[d]);
    atomicAdd(&o2[(long long)b * S + h2[d]], v * s2[d]);
    atomicAdd(&o3[(long long)b * S + h3[d]], v * s3[d]);
}

__global__ void sketch_mul_kernel(const float* __restrict__ a,
                                  const float* __restrict__ b,
                                  const float* __restrict__ c,
                                  float* __restrict__ out, long long n)
{
    const long long idx = (long long)blockIdx.x * 256 + threadIdx.x;
    if (idx < n) out[idx] = a[idx] * b[idx] * c[idx];
}

// ---------------------------------------------------------------------------
extern "C" void kernel_launch(void* const* d_in, const int* in_sizes, int n_in,
                              void* d_out, int out_size, void* d_ws, size_t ws_size,
                              hipStream_t stream)
{
    (void)in_sizes; (void)n_in; (void)out_size; (void)ws_size;

    constexpr int Bn = 64, Nn = 196, Dd = 384, Sk = 4096;
    constexpr int TRIc = Dd * (Dd + 1) / 2;   // 73920
    constexpr int H2 = 1024, O2 = 512, H3 = 1024, O3 = 512;

    const float* tokens = (const float*)d_in[0];
    const float* graph  = (const float*)d_in[1];
    const int*   hash1  = (const int*)d_in[2];
    const int*   hash2  = (const int*)d_in[3];
    const int*   hash3  = (const int*)d_in[4];
    const float* sign1  = (const float*)d_in[5];
    const float* sign2  = (const float*)d_in[6];
    const float* sign3  = (const float*)d_in[7];
    const float* w1_2   = (const float*)d_in[8];
    const float* b1_2   = (const float*)d_in[9];
    const float* gam2   = (const float*)d_in[10];
    const float* bet2   = (const float*)d_in[11];
    const float* rm2    = (const float*)d_in[12];
    const float* rv2    = (const float*)d_in[13];
    const float* w2_2   = (const float*)d_in[14];
    const float* b2_2   = (const float*)d_in[15];
    const float* w1_3   = (const float*)d_in[16];
    const float* b1_3   = (const float*)d_in[17];
    const float* gam3   = (const float*)d_in[18];
    const float* bet3   = (const float*)d_in[19];
    const float* rm3    = (const float*)d_in[20];
    const float* rv3    = (const float*)d_in[21];
    const float* w2_3   = (const float*)d_in[22];
    const float* b2_3   = (const float*)d_in[23];
    float* out = (float*)d_out;

    // workspace carve-up (floats)
    float* p = (float*)d_ws;
    auto alloc = [&](long long n) { float* r = p; p += n; return r; };
    float* invs   = alloc((long long)Bn * Nn);
    float* rowsum = alloc((long long)Bn * Nn);
    float* tr_w   = alloc(Bn);
    float* mu     = alloc((long long)Bn * Dd);
    float* wc     = alloc((long long)Bn * Dd);
    float* W      = alloc((long long)Bn * Nn * Nn);
    float* Xc     = alloc((long long)Bn * Nn * Dd);
    float* WX     = alloc((long long)Bn * Nn * Dd);
    float* ZA     = alloc((long long)Bn * Dd * Dd);   // M2, then Z
    float* YA     = alloc((long long)Bn * Dd * Dd);
    float* YB     = alloc((long long)Bn * Dd * Dd);
    float* ZB     = alloc((long long)Bn * Dd * Dd);
    float* P1     = alloc((long long)Bn * Dd * Dd);
    float* P2     = alloc((long long)Bn * Dd * Dd);
    float* trM2   = alloc(Bn);
    float* M2vec  = alloc((long long)Bn * TRIc);
    float* h2     = alloc((long long)Bn * H2);
    float* s123   = alloc(3LL * Bn * Sk);
    float* sk     = alloc((long long)Bn * Sk);
    float* h3     = alloc((long long)Bn * H3);
    float* s1 = s123, *s2 = s123 + (long long)Bn * Sk, *s3 = s123 + 2LL * Bn * Sk;

    auto gemm = [&](const float* A, const float* Bm, float* C, const float* bias,
                    int M, int N, int K, int lda, int ldb, int ldc,
                    long long sA, long long sB, long long sC,
                    int batch, int transA, float alpha) {
        dim3 grid((N + 127) / 128, (M + 127) / 128, batch);
        if (transA)
            gemm_f32_wmma<1><<<grid, 256, 0, stream>>>(A, Bm, C, bias, M, N, K,
                                                       lda, ldb, ldc, sA, sB, sC, alpha);
        else
            gemm_f32_wmma<0><<<grid, 256, 0, stream>>>(A, Bm, C, bias, M, N, K,
                                                       lda, ldb, ldc, sA, sB, sC, alpha);
    };

    // 1) graph normalization
    deg_kernel<<<Bn * Nn, 256, 0, stream>>>(graph, invs, Nn);
    w_rowsum_kernel<<<Bn * Nn, 256, 0, stream>>>(graph, invs, W, rowsum, Nn);
    trace_kernel<<<Bn, 256, 0, stream>>>(W, tr_w, Nn);
    mu_wc_kernel<<<Bn, Dd, 0, stream>>>(tokens, rowsum, tr_w, mu, wc, Nn, Dd);
    xc_kernel<<<Bn * Nn, Dd, 0, stream>>>(tokens, mu, Xc, Nn, Dd);

    // 2) WX = W @ Xc ; M2 = Xc^T @ WX   (batched WMMA GEMMs)
    gemm(W, Xc, WX, nullptr, Nn, Dd, Nn, Nn, Dd, Dd,
         (long long)Nn * Nn, (long long)Nn * Dd, (long long)Nn * Dd, Bn, 0, 1.0f);
    gemm(Xc, WX, ZA, nullptr, Dd, Dd, Nn, Dd, Dd, Dd,
         (long long)Nn * Dd, (long long)Nn * Dd, (long long)Dd * Dd, Bn, 1, 1.0f);

    // 3) Newton-Schulz inverse sqrt
    trace_kernel<<<Bn, 256, 0, stream>>>(ZA, trM2, Dd);
    {
        dim3 g((Dd * Dd + 255) / 256, Bn);
        ns_init_kernel<<<g, 256, 0, stream>>>(ZA, YA, trM2, Dd);
    }
    {
        const long long sM = (long long)Dd * Dd;
        float *Y = YA, *Z = ZA, *Yn = YB, *Zn = ZB;
        dim3 ge((Dd * Dd + 255) / 256, Bn);
        for (int it = 0; it < 5; ++it) {
            gemm(Z, Y, P1, nullptr, Dd, Dd, Dd, Dd, Dd, Dd, sM, sM, sM, Bn, 0, 1.0f);
            gemm(Y, Z, P2, nullptr, Dd, Dd, Dd, Dd, Dd, Dd, sM, sM, sM, Bn, 0, 1.0f);
            neg3i_kernel<<<ge, 256, 0, stream>>>(P1, Dd);
            neg3i_kernel<<<ge, 256, 0, stream>>>(P2, Dd);
            gemm(Y, P1, Yn, nullptr, Dd, Dd, Dd, Dd, Dd, Dd, sM, sM, sM, Bn, 0, 0.5f);
            gemm(P2, Z, Zn, nullptr, Dd, Dd, Dd, Dd, Dd, Dd, sM, sM, sM, Bn, 0, 0.5f);
            float* t;
            t = Y; Y = Yn; Yn = t;
            t = Z; Z = Zn; Zn = t;
        }
        dim3 gt((TRIc + 255) / 256, Bn);
        triu_kernel<<<gt, 256, 0, stream>>>(Y, trM2, M2vec, Dd, TRIc);
    }

    // 5) second head MLP
    gemm(M2vec, w1_2, h2, b1_2, Bn, H2, TRIc, TRIc, H2, H2, 0, 0, 0, 1, 0, 1.0f);
    bn_gelu_kernel<<<(int)(((long long)Bn * H2 + 255) / 256), 256, 0, stream>>>(
        h2, gam2, bet2, rm2, rv2, H2, (long long)Bn * H2);
    gemm(h2, w2_2, out, b2_2, Bn, O2, H2, H2, O2, 1024, 0, 0, 0, 1, 0, 1.0f);

    // 6) count sketch
    zero_kernel<<<(int)((3LL * Bn * Sk + 255) / 256), 256, 0, stream>>>(s123, 3LL * Bn * Sk);
    sketch_scatter_kernel<<<Bn, Dd, 0, stream>>>(wc, hash1, hash2, hash3,
                                                 sign1, sign2, sign3,
                                                 s1, s2, s3, Dd, Sk);
    sketch_mul_kernel<<<(int)(((long long)Bn * Sk + 255) / 256), 256, 0, stream>>>(
        s1, s2, s3, sk, (long long)Bn * Sk);

    // 7) third head MLP
    gemm(sk, w1_3, h3, b1_3, Bn, H3, Sk, Sk, H3, H3, 0, 0, 0, 1, 0, 1.0f);
    bn_gelu_kernel<<<(int)(((long long)Bn * H3 + 255) / 256), 256, 0, stream>>>(
        h3, gam3, bet3, rm3, rv3, H3, (long long)Bn * H3);
    gemm(h3, w2_3, out + O2, b2_3, Bn, O3, H3, H3, O3, 1024, 0, 0, 0, 1, 0, 1.0f);
}